// GCN_layer_54554674594287
// MI455X (gfx1250) — compile-verified
//
#include <hip/hip_runtime.h>

// ---------------------------------------------------------------------------
// GCN layer for MI455X (gfx1250, wave32):
//   WT      = W^T                           (tiny transpose, 256 KB)
//   support = X @ W                         (fp32 V_WMMA_F32_16X16X4_F32)
//   out     = bias + SpMM(edges, support)   (gather + global_atomic_add_f32)
// ---------------------------------------------------------------------------

typedef __attribute__((ext_vector_type(2))) float v2f;
typedef __attribute__((ext_vector_type(8))) float v8f;

#define N_NODES 100000
#define DIM     256
#define N_EDGES 3200000

// ---------------------------------------------------------------------------
// Kernel 0: WT[n][k] = W[k][n].  256 blocks (one per k-row), 256 threads (n).
// Coalesced reads; writes scatter with stride 1 KB but total is only 256 KB.
// ---------------------------------------------------------------------------
__global__ __launch_bounds__(256) void transpose_w_kernel(
    const float* __restrict__ W, float* __restrict__ WT) {
  const int k = blockIdx.x;
  const int n = threadIdx.x;
  WT[(size_t)n * DIM + k] = W[(size_t)k * DIM + n];
}

// ---------------------------------------------------------------------------
// Kernel 1: out[r, c] = bias[c]   (float4 stores; seeds the accumulator)
// ---------------------------------------------------------------------------
__global__ __launch_bounds__(256) void init_bias_kernel(
    float* __restrict__ out, const float* __restrict__ bias) {
  size_t idx = (size_t)blockIdx.x * 256 + threadIdx.x;   // float4 index
  int c4 = (int)(idx & 63);                              // 64 float4 per row
  float4 b = ((const float4*)bias)[c4];
  ((float4*)out)[idx] = b;
}

// ---------------------------------------------------------------------------
// Kernel 2: S = X @ W with fp32 WMMA, reading B from WT.
//   Block = 256 threads = 8 waves; block computes 32 rows x 256 cols.
//   Waves 0-3 -> rows [0,16), waves 4-7 -> rows [16,32); each wave owns
//   4 N-tiles (64 cols).  K-loop steps by 4: 5 x b64 loads feed 4 WMMAs.
//
// Fragment layouts (ISA 7.12.2, wave32):
//   A 16x4 f32 : lanes 0-15 -> M=lane, VGPRs = K={0,1}; lanes 16-31 -> K={2,3}
//   B 4x16 f32 : VGPR0 = K=0 (lanes 0-15) / K=2 (lanes 16-31); VGPR1 = K=1/K=3
//     -> with WT[n][k], fragment = contiguous b64 at WT[n][k0+khalf]
//   C/D 16x16  : VGPR v = row (v + 8*(lane>>4)), col = lane&15
// ---------------------------------------------------------------------------
__global__ __launch_bounds__(256) void gemm_wmma_kernel(
    const float* __restrict__ X, const float* __restrict__ WT,
    float* __restrict__ S) {
  const int lane  = threadIdx.x & 31;
  const int wave  = threadIdx.x >> 5;
  const int row0  = blockIdx.x * 32 + (wave >> 2) * 16;  // this wave's 16 rows
  const int n0    = (wave & 3) * 64;                     // this wave's 64 cols
  const int l15   = lane & 15;
  const int khalf = (lane >> 4) * 2;   // 0 for lanes 0-15, 2 for lanes 16-31

  v8f acc0 = {}, acc1 = {}, acc2 = {}, acc3 = {};

  const float* xrow = X  + (size_t)(row0 + l15) * DIM + khalf;
  const float* wt0  = WT + (size_t)(n0 + l15) * DIM + khalf;        // tile 0
  const float* wt1  = wt0 + (size_t)16 * DIM;                       // tile 1
  const float* wt2  = wt0 + (size_t)32 * DIM;                       // tile 2
  const float* wt3  = wt0 + (size_t)48 * DIM;                       // tile 3

  #pragma unroll 4
  for (int k0 = 0; k0 < DIM; k0 += 4) {
    const float2 ax = *(const float2*)(xrow + k0);
    const float2 b0 = *(const float2*)(wt0 + k0);
    const float2 b1 = *(const float2*)(wt1 + k0);
    const float2 b2 = *(const float2*)(wt2 + k0);
    const float2 b3 = *(const float2*)(wt3 + k0);
    v2f A;  A.x  = ax.x; A.y  = ax.y;
    v2f B0; B0.x = b0.x; B0.y = b0.y;
    v2f B1; B1.x = b1.x; B1.y = b1.y;
    v2f B2; B2.x = b2.x; B2.y = b2.y;
    v2f B3; B3.x = b3.x; B3.y = b3.y;

    acc0 = __builtin_amdgcn_wmma_f32_16x16x4_f32(
        false, A, false, B0, (short)0, acc0, false, false);
    acc1 = __builtin_amdgcn_wmma_f32_16x16x4_f32(
        false, A, false, B1, (short)0, acc1, false, false);
    acc2 = __builtin_amdgcn_wmma_f32_16x16x4_f32(
        false, A, false, B2, (short)0, acc2, false, false);
    acc3 = __builtin_amdgcn_wmma_f32_16x16x4_f32(
        false, A, false, B3, (short)0, acc3, false, false);
  }

  // Store C/D tiles: row = v + 8*(lane>>4), col = n0 + l15 + 16*t
  const int mhi = (lane >> 4) * 8;
  float* sp = S + (size_t)(row0 + mhi) * DIM + n0 + l15;
  #pragma unroll
  for (int v = 0; v < 8; ++v) {
    sp[(size_t)v * DIM]      = acc0[v];
    sp[(size_t)v * DIM + 16] = acc1[v];
    sp[(size_t)v * DIM + 32] = acc2[v];
    sp[(size_t)v * DIM + 48] = acc3[v];
  }
}

// ---------------------------------------------------------------------------
// Kernel 3: edge-parallel SpMM scatter.
//   One wave per edge; lane owns 8 consecutive columns.
//   Gather: two b128 loads of support[col] (L2-resident, 100 MB < 192 MB L2).
//   Scatter: 8 non-returning global_atomic_add_f32 at device scope
//   (fire-and-forget, STOREcnt tracked; s_endpgm's implicit wait drains them).
// ---------------------------------------------------------------------------
__device__ __forceinline__ void atomic_add_f32_dev(float* p, float v, int) {
  __hip_atomic_fetch_add(p, v, __ATOMIC_RELAXED, __HIP_MEMORY_SCOPE_AGENT);
}

__global__ __launch_bounds__(256) void spmm_edge_kernel(
    const float* __restrict__ S, const int* __restrict__ erow,
    const int* __restrict__ ecol, const float* __restrict__ eval,
    float* __restrict__ out) {
  const int lane = threadIdx.x & 31;
  const int wave = threadIdx.x >> 5;
  const size_t e = (size_t)blockIdx.x * 8 + wave;

  const int   r = erow[e];
  const int   c = ecol[e];
  const float v = eval[e];

  const float4* src = (const float4*)(S + (size_t)c * DIM) + lane * 2;
  const float4 g0 = src[0];
  const float4 g1 = src[1];

  float* dst = out + (size_t)r * DIM + lane * 8;
  const float a0 = v * g0.x, a1 = v * g0.y, a2 = v * g0.z, a3 = v * g0.w;
  const float a4 = v * g1.x, a5 = v * g1.y, a6 = v * g1.z, a7 = v * g1.w;

  // Pin the hardware FP atomic (ISA §15.18.3 op 86), device scope, no return.
  asm volatile("global_atomic_add_f32 %0, %1, off scope:SCOPE_DEV"
               :: "v"(dst), "v"(a0) : "memory");
  asm volatile("global_atomic_add_f32 %0, %1, off offset:4 scope:SCOPE_DEV"
               :: "v"(dst), "v"(a1) : "memory");
  asm volatile("global_atomic_add_f32 %0, %1, off offset:8 scope:SCOPE_DEV"
               :: "v"(dst), "v"(a2) : "memory");
  asm volatile("global_atomic_add_f32 %0, %1, off offset:12 scope:SCOPE_DEV"
               :: "v"(dst), "v"(a3) : "memory");
  asm volatile("global_atomic_add_f32 %0, %1, off offset:16 scope:SCOPE_DEV"
               :: "v"(dst), "v"(a4) : "memory");
  asm volatile("global_atomic_add_f32 %0, %1, off offset:20 scope:SCOPE_DEV"
               :: "v"(dst), "v"(a5) : "memory");
  asm volatile("global_atomic_add_f32 %0, %1, off offset:24 scope:SCOPE_DEV"
               :: "v"(dst), "v"(a6) : "memory");
  asm volatile("global_atomic_add_f32 %0, %1, off offset:28 scope:SCOPE_DEV"
               :: "v"(dst), "v"(a7) : "memory");
}

// ---------------------------------------------------------------------------
// Launch: WT transpose -> init(out=bias) -> GEMM(supp in d_ws) -> scatter.
// d_ws layout: [0, 256KB) = WT,  [256KB, 256KB + 102.4MB) = support.
// ---------------------------------------------------------------------------
extern "C" void kernel_launch(void* const* d_in, const int* in_sizes, int n_in,
                              void* d_out, int out_size, void* d_ws, size_t ws_size,
                              hipStream_t stream) {
  const float* X    = (const float*)d_in[0];
  const float* W    = (const float*)d_in[1];
  const float* bias = (const float*)d_in[2];
  const int*   er   = (const int*)d_in[3];
  const int*   ec   = (const int*)d_in[4];
  const float* ev   = (const float*)d_in[5];
  float*       out  = (float*)d_out;

  float* WT   = (float*)d_ws;            // [DIM, DIM]
  float* supp = WT + (size_t)DIM * DIM;  // [N_NODES, DIM]

  transpose_w_kernel<<<DIM, DIM, 0, stream>>>(W, WT);
  init_bias_kernel<<<(N_NODES * (DIM / 4)) / 256, 256, 0, stream>>>(out, bias);
  gemm_wmma_kernel<<<N_NODES / 32, 256, 0, stream>>>(X, WT, supp);
  spmm_edge_kernel<<<N_EDGES / 8, 256, 0, stream>>>(supp, er, ec, ev, out);
}